// GATLayer_77893526880541
// MI455X (gfx1250) — compile-verified
//
#include <hip/hip_runtime.h>

#define NROWS 8192
#define FIN   512
#define FOUT  256
#define CC    768            // 3*FOUT
#define K1    1536           // 3*FIN
#define ALPHA_LRELU 0.2f
#define NEG_BIG (-9.0e15f)

typedef __attribute__((ext_vector_type(2))) float v2f;
typedef __attribute__((ext_vector_type(8))) float v8f;
typedef unsigned int u32x4 __attribute__((ext_vector_type(4)));
typedef int i32x8 __attribute__((ext_vector_type(8)));
typedef int i32x4 __attribute__((ext_vector_type(4)));

__device__ __forceinline__ v8f wmma_f32_k4(v2f a, v2f b, v8f c) {
  // V_WMMA_F32_16X16X4_F32 : D = A(16x4) * B(4x16) + C(16x16)
  return __builtin_amdgcn_wmma_f32_16x16x4_f32(
      /*neg_a=*/false, a, /*neg_b=*/false, b,
      /*c_mod=*/(short)0, c, /*reuse_a=*/false, /*reuse_b=*/false);
}

// ---------------------------------------------------------------------------
// TDM: DMA a 16x128 f32 tile (row stride 8192 elements) from an 8192x8192
// tensor into LDS. D# built per CDNA5 ISA 8.3/8.4; groups 2/3 zero (2D).
// This toolchain exposes the 6-arg builtin (extra int32x8 group + cpol).
// ---------------------------------------------------------------------------
__device__ __forceinline__ void tdm_load_tile_16x128(unsigned lds_addr,
                                                     const float* g_tile) {
  unsigned long long ga = (unsigned long long)(size_t)g_tile;
  u32x4 g0;
  g0[0] = 1u;                                            // count=1, flags 0
  g0[1] = lds_addr;                                      // lds_addr[31:0]
  g0[2] = (unsigned)(ga & 0xFFFFFFFFu);                  // global_addr lo
  g0[3] = (unsigned)((ga >> 32) & 0x01FFFFFFu)           // global_addr hi
          | 0x80000000u;                                 // type=2 ("image")
  i32x8 g1;
  g1[0] = (int)(2u << 16);                               // data_size = 4B
  g1[1] = (int)((8192u & 0xFFFFu) << 16);                // tensor_dim0 lo16
  g1[2] = (int)(((8192u >> 16) & 0xFFFFu)                // tensor_dim0 hi16
          | ((8192u & 0xFFFFu) << 16));                  // tensor_dim1 lo16
  g1[3] = (int)(((8192u >> 16) & 0xFFFFu)                // tensor_dim1 hi16
          | (128u << 16));                               // tile_dim0 = 128
  g1[4] = 16;                                            // tile_dim1=16, tile_dim2=0
  g1[5] = 8192;                                          // dim0_stride lo32
  g1[6] = 0;                                             // stride hi / dim1_stride
  g1[7] = 0;
  i32x4 z4 = {};
  i32x8 z8 = {};
  __builtin_amdgcn_tensor_load_to_lds(g0, g1, z4, z4, z8, 0);
}

__device__ __forceinline__ unsigned lds_addr_of(const void* p) {
  // __shared__ pointer -> flat: {shared_aperture, offset}; LDS addr = low 32b
  return (unsigned)(size_t)p;
}

// ---------------------------------------------------------------------------
// Kernel 1: pack W[FIN, FOUT, 3] into block-circulant Bp[K1, CC]
//   Bp[i*3+j, o*3+k] = W[i, o, (k-j) mod 3]
// ---------------------------------------------------------------------------
__global__ __launch_bounds__(256) void pack_w_kernel(const float* __restrict__ W,
                                                     float* __restrict__ Bp) {
  int idx = blockIdx.x * 256 + threadIdx.x;
  if (idx >= K1 * CC) return;
  int kk = idx / CC;          // i*3 + j
  int c  = idx % CC;          // o*3 + k2
  int i = kk / 3, j = kk % 3;
  int o = c / 3,  k2 = c % 3;
  int ks = k2 - j; if (ks < 0) ks += 3;
  Bp[idx] = W[((size_t)i * FOUT + o) * 3 + ks];
}

// ---------------------------------------------------------------------------
// Kernel 2: H[8192, 768] = X[8192, 1536] @ Bp[1536, 768]  (f32 WMMA)
// ---------------------------------------------------------------------------
__global__ __launch_bounds__(256) void gemm1_kernel(const float* __restrict__ X,
                                                    const float* __restrict__ Bp,
                                                    float* __restrict__ H) {
  const int i0   = blockIdx.x * 16;
  const int lane = threadIdx.x & 31;
  const int wave = threadIdx.x >> 5;
  const int c0   = wave * 96;
  const int hi   = lane >> 4;
  const int arow = lane & 15;

  v8f acc[6];
  v8f zero = {};
  #pragma unroll
  for (int t = 0; t < 6; ++t) acc[t] = zero;

  const float* xrow = X + (size_t)(i0 + arow) * K1;

  for (int kk = 0; kk < K1; kk += 4) {
    v2f a;
    a.x = xrow[kk + 2 * hi + 0];
    a.y = xrow[kk + 2 * hi + 1];
    #pragma unroll
    for (int t = 0; t < 6; ++t) {
      int col = c0 + t * 16 + arow;
      v2f b;
      b.x = Bp[(size_t)(kk + hi)     * CC + col];
      b.y = Bp[(size_t)(kk + 2 + hi) * CC + col];
      acc[t] = wmma_f32_k4(a, b, acc[t]);
    }
  }

  #pragma unroll
  for (int t = 0; t < 6; ++t) {
    int col = c0 + t * 16 + arow;
    #pragma unroll
    for (int r = 0; r < 8; ++r) {
      int row = i0 + r + 8 * hi;
      H[(size_t)row * CC + col] = acc[t][r];
    }
  }
}

// ---------------------------------------------------------------------------
// Kernel 3: e_self[n] = H[n,:]·a_self ; e_nei[n] = H[n,:]·a_neighs
// ---------------------------------------------------------------------------
__global__ __launch_bounds__(256) void e_kernel(const float* __restrict__ H,
                                                const float* __restrict__ a_self,
                                                const float* __restrict__ a_nei,
                                                float* __restrict__ e_self,
                                                float* __restrict__ e_nei) {
  __shared__ float ss[256];
  __shared__ float sn[256];
  const int n = blockIdx.x;
  const int t = threadIdx.x;
  const float* hrow = H + (size_t)n * CC;
  float ps = 0.f, pn = 0.f;
  #pragma unroll
  for (int u = 0; u < 3; ++u) {
    int c = t * 3 + u;
    float hv = hrow[c];
    ps += hv * a_self[c];
    pn += hv * a_nei[c];
  }
  ss[t] = ps; sn[t] = pn;
  __syncthreads();
  for (int s = 128; s > 0; s >>= 1) {
    if (t < s) { ss[t] += ss[t + s]; sn[t] += sn[t + s]; }
    __syncthreads();
  }
  if (t == 0) { e_self[n] = ss[0]; e_nei[n] = sn[0]; }
}

// ---------------------------------------------------------------------------
// Kernel 4: per-row online softmax stats over the (recomputed) score row.
// ---------------------------------------------------------------------------
__global__ __launch_bounds__(256) void stats_kernel(const float* __restrict__ Mm,
                                                    const float* __restrict__ adj,
                                                    const float* __restrict__ e_self,
                                                    const float* __restrict__ e_nei,
                                                    float* __restrict__ rowmax,
                                                    float* __restrict__ rowsum) {
  __shared__ float sm[256];
  __shared__ float sl[256];
  const int i = blockIdx.x;
  const int t = threadIdx.x;
  const float es = e_self[i];
  const float* mrow = Mm  + (size_t)i * NROWS;
  const float* arow = adj + (size_t)i * NROWS;

  float m = -__builtin_inff(), l = 0.f;
  for (int j = t; j < NROWS; j += 256) {
    float v = (es + e_nei[j]) * mrow[j];
    v = v > 0.f ? v : ALPHA_LRELU * v;
    float s = arow[j] > 0.f ? v : NEG_BIG;
    float mn = fmaxf(m, s);
    l = l * __expf(m - mn) + __expf(s - mn);
    m = mn;
  }
  sm[t] = m; sl[t] = l;
  __syncthreads();
  for (int s = 128; s > 0; s >>= 1) {
    if (t < s) {
      float m2 = sm[t + s], l2 = sl[t + s];
      float mn = fmaxf(sm[t], m2);
      sl[t] = sl[t] * __expf(sm[t] - mn) + l2 * __expf(m2 - mn);
      sm[t] = mn;
    }
    __syncthreads();
  }
  if (t == 0) { rowmax[i] = sm[0]; rowsum[i] = sl[0]; }
}

// ---------------------------------------------------------------------------
// Kernel 5: out = elu( softmax(scores) @ H ), flash-style with WMMA.
// TDM double-buffers 16x128 tiles of M and adj into LDS (TENSORcnt pipelined
// against the WMMA phase); P = exp(score - rowmax) built in LDS; 8 waves x
// 6 C-tiles of f32 WMMA against L2-resident H; epilogue = 1/rowsum + ELU.
// ---------------------------------------------------------------------------
__global__ __launch_bounds__(256) void attn_gemm_kernel(
    const float* __restrict__ Mm, const float* __restrict__ adj,
    const float* __restrict__ H,
    const float* __restrict__ e_self, const float* __restrict__ e_nei,
    const float* __restrict__ rowmax, const float* __restrict__ rowsum,
    float* __restrict__ out) {
  __shared__ float Mb[2][16 * 128];
  __shared__ float Ab[2][16 * 128];
  __shared__ float P[16 * 132];      // padded stride to spread LDS banks
  __shared__ float s_es[16], s_rmax[16], s_rsinv[16];

  const int i0   = blockIdx.x * 16;
  const int t    = threadIdx.x;
  const int lane = t & 31;
  const int wave = t >> 5;
  const int c0   = wave * 96;
  const int hi   = lane >> 4;
  const int arow = lane & 15;

  if (t < 16) {
    s_es[t]    = e_self[i0 + t];
    s_rmax[t]  = rowmax[i0 + t];
    s_rsinv[t] = 1.0f / rowsum[i0 + t];
  }

  // Prime the pipeline: chunk 0 -> buffer 0.
  if (wave == 0) {
    tdm_load_tile_16x128(lds_addr_of(&Mb[0][0]), Mm  + (size_t)i0 * NROWS);
    tdm_load_tile_16x128(lds_addr_of(&Ab[0][0]), adj + (size_t)i0 * NROWS);
  }

  v8f acc[6];
  v8f zero = {};
  #pragma unroll
  for (int q = 0; q < 6; ++q) acc[q] = zero;

  const int cj    = t & 127;         // column within chunk
  const int rbase = (t >> 7) * 8;    // rows 0-7 or 8-15

  for (int j0 = 0; j0 < NROWS; j0 += 128) {
    const int buf = (j0 >> 7) & 1;
    if (wave == 0) {
      if (j0 + 128 < NROWS) {
        // Issue next chunk into the other buffer, then wait until the two
        // loads of the CURRENT chunk retired (2 newer ones stay in flight).
        const int nb = buf ^ 1;
        tdm_load_tile_16x128(lds_addr_of(&Mb[nb][0]),
                             Mm  + (size_t)i0 * NROWS + (j0 + 128));
        tdm_load_tile_16x128(lds_addr_of(&Ab[nb][0]),
                             adj + (size_t)i0 * NROWS + (j0 + 128));
        __builtin_amdgcn_s_wait_tensorcnt(2);
      } else {
        __builtin_amdgcn_s_wait_tensorcnt(0);
      }
    }
    __syncthreads();                 // tile (and round-0 s_* fills) visible

    const float en = e_nei[j0 + cj];
    const float* mb = &Mb[buf][0];
    const float* ab = &Ab[buf][0];
    #pragma unroll
    for (int r = 0; r < 8; ++r) {
      int row = rbase + r;
      float v = (s_es[row] + en) * mb[row * 128 + cj];
      v = v > 0.f ? v : ALPHA_LRELU * v;
      float s = ab[row * 128 + cj] > 0.f ? v : NEG_BIG;
      P[row * 132 + cj] = __expf(s - s_rmax[row]);
    }
    __syncthreads();                 // P ready

    for (int jj = 0; jj < 128; jj += 4) {
      v2f a;
      a.x = P[arow * 132 + jj + 2 * hi];
      a.y = P[arow * 132 + jj + 2 * hi + 1];
      const float* hb0 = H + (size_t)(j0 + jj + hi)     * CC;
      const float* hb1 = H + (size_t)(j0 + jj + 2 + hi) * CC;
      #pragma unroll
      for (int tt = 0; tt < 6; ++tt) {
        int col = c0 + tt * 16 + arow;
        v2f b;
        b.x = hb0[col];
        b.y = hb1[col];
        acc[tt] = wmma_f32_k4(a, b, acc[tt]);
      }
    }
    __syncthreads();                 // P and Mb/Ab[buf] fully consumed
  }

  #pragma unroll
  for (int tt = 0; tt < 6; ++tt) {
    int col = c0 + tt * 16 + arow;
    #pragma unroll
    for (int r = 0; r < 8; ++r) {
      int rl = r + 8 * hi;
      float v = acc[tt][r] * s_rsinv[rl];     // softmax normalize
      v = v > 0.f ? v : (__expf(v) - 1.0f);   // elu
      out[(size_t)(i0 + rl) * CC + col] = v;
    }
  }
}

// ---------------------------------------------------------------------------
extern "C" void kernel_launch(void* const* d_in, const int* in_sizes, int n_in,
                              void* d_out, int out_size, void* d_ws, size_t ws_size,
                              hipStream_t stream) {
  const float* x      = (const float*)d_in[0];  // [N, FIN, 3]
  const float* W      = (const float*)d_in[1];  // [FIN, FOUT, 3]
  const float* a_self = (const float*)d_in[2];  // [768, 1]
  const float* a_nei  = (const float*)d_in[3];  // [768, 1]
  const float* adj    = (const float*)d_in[4];  // [N, N]
  const float* Mm     = (const float*)d_in[5];  // [N, N]
  float* out = (float*)d_out;                   // [N, FOUT, 3] = [N, 768]

  float* ws     = (float*)d_ws;
  float* Bp     = ws;                              // K1*CC
  float* H      = Bp + (size_t)K1 * CC;            // NROWS*CC
  float* e_self = H  + (size_t)NROWS * CC;         // NROWS
  float* e_nei  = e_self + NROWS;                  // NROWS
  float* rowmax = e_nei  + NROWS;                  // NROWS
  float* rowsum = rowmax + NROWS;                  // NROWS

  pack_w_kernel<<<(K1 * CC + 255) / 256, 256, 0, stream>>>(W, Bp);
  gemm1_kernel<<<NROWS / 16, 256, 0, stream>>>(x, Bp, H);
  e_kernel<<<NROWS, 256, 0, stream>>>(H, a_self, a_nei, e_self, e_nei);
  stats_kernel<<<NROWS, 256, 0, stream>>>(Mm, adj, e_self, e_nei, rowmax, rowsum);
  attn_gemm_kernel<<<NROWS / 16, 256, 0, stream>>>(Mm, adj, H, e_self, e_nei,
                                                   rowmax, rowsum, out);
}